// EdgePredictorVN_83786222011214
// MI455X (gfx1250) — compile-verified
//
#include <hip/hip_runtime.h>
#include <math.h>

// ---------------- problem constants (from the reference) ----------------
#define N_LIG   8192
#define MOLS    256
#define NPM     32          // nodes per molecule
#define EPM     496         // triu(32, k=1) edges per molecule
#define EP      126976      // 256 * 496
#define IN_SCA  256
#define IN_VEC  64
#define KEY_SCA 128
#define HEADS   16

typedef float v2f __attribute__((ext_vector_type(2)));
typedef float v8f __attribute__((ext_vector_type(8)));

// ---- Tensor Data Mover availability / arity detection ----
#if __has_builtin(__builtin_amdgcn_tensor_load_to_lds) && \
    __has_builtin(__builtin_amdgcn_s_wait_tensorcnt)
#define HAVE_TDM 1
#if __has_include(<hip/amd_detail/amd_gfx1250_TDM.h>)
#define TDM_6ARG 1   // amdgpu-toolchain (clang-23): 6-arg builtin
#endif
#endif

typedef unsigned int u32x4 __attribute__((ext_vector_type(4)));
typedef int i32x4 __attribute__((ext_vector_type(4)));
typedef int i32x8 __attribute__((ext_vector_type(8)));

struct GVParams {
  const float* wv1;   // (dim_hid, in_v)
  const float* wsca;  // (out_s, in_s + dim_hid)
  const float* wv2;   // (out_v, dim_hid)
  const float* wg;    // (out_v, out_s)
  const float* bg;    // (out_v,)
};

// D[m][n] = sum_k A[m][k] * B[n][k] + C  -- both operands fed as [outer=16][K=4],
// outer dim on lanes 0-15, K split (2 VGPRs) x (lane-half).
__device__ __forceinline__ v8f wmma4(v2f a, v2f b, v8f c) {
  return __builtin_amdgcn_wmma_f32_16x16x4_f32(false, a, false, b, (short)0, c,
                                               false, false);
}

__device__ __forceinline__ float wave_sum(float v) {
#pragma unroll
  for (int off = 16; off; off >>= 1) v += __shfl_xor(v, off, 32);
  return v;
}

#ifdef HAVE_TDM
// One 1-D TDM descriptor: copy nelem f32 (contiguous) global -> LDS.
// D# layout per CDNA5 ISA.
__device__ __forceinline__ void tdm_load_1d(unsigned int lds_addr,
                                            const void* gptr,
                                            unsigned int nelem) {
  unsigned long long ga = (unsigned long long)(uintptr_t)gptr;
  u32x4 g0;
  g0[0] = 1u;                                   // count=1 (valid), user mode
  g0[1] = lds_addr;                             // lds_addr (bytes)
  g0[2] = (unsigned int)ga;                     // global_addr[31:0]
  g0[3] = (unsigned int)(ga >> 32) | (2u << 30);// global_addr[56:32] | type=2
  i32x8 g1;
  g1[0] = (int)(2u << 16);                      // data_size = 4 bytes
  g1[1] = (int)((nelem & 0xFFFFu) << 16);       // tensor_dim0[15:0] @ bits63:48
  g1[2] = (int)((nelem >> 16) | (1u << 16));    // tensor_dim0[31:16] | tensor_dim1=1
  g1[3] = (int)((nelem & 0xFFFFu) << 16);       // tile_dim0 @ bits127:112
  g1[4] = 1;                                    // tile_dim1=1, tile_dim2=0
  g1[5] = (int)nelem;                           // tensor_dim0_stride[31:0]
  g1[6] = 0;                                    // stride hi / dim1_stride lo
  g1[7] = 0;                                    // dim1_stride hi
  i32x4 z4 = {0, 0, 0, 0};
#ifdef TDM_6ARG
  i32x8 z8 = {0, 0, 0, 0, 0, 0, 0, 0};
  __builtin_amdgcn_tensor_load_to_lds(g0, g1, z4, z4, z8, 0);
#else
  __builtin_amdgcn_tensor_load_to_lds(g0, g1, z4, z4, 0);
#endif
}
#endif

// =====================================================================
// Kernel 1: GVLinear for q, k, v.  One wave per 16-node tile.
// =====================================================================
__global__ __launch_bounds__(32) void gv_qkv_kernel(
    const float* __restrict__ hsca, const float* __restrict__ hvec,
    const int* __restrict__ idxl,
    GVParams pq, GVParams pk, GVParams pv,
    float* __restrict__ qs, float* __restrict__ ksp, float* __restrict__ vsp,
    float* __restrict__ qv, float* __restrict__ kvp, float* __restrict__ vvp) {
  __shared__ float vhL[3][16][64];
  __shared__ float vnL[16][64];
  __shared__ float osL[16][IN_SCA];
  __shared__ float gtL[16][64];

  const int tile = blockIdx.x;
  const int lane = threadIdx.x & 31;
  const int m = lane & 15;   // node row within tile  (== B n-index)
  const int h = lane >> 4;   // K-half selector

  const int nid = idxl[tile * 16 + m];
  const float* sca_row = hsca + (size_t)nid * IN_SCA;
  const float* vec_row = hvec + (size_t)nid * (IN_VEC * 3);

  GVParams P[3] = {pq, pk, pv};
  float* outSg[3] = {qs, ksp, vsp};
  float* outVg[3] = {qv, kvp, vvp};
  const int OSd[3] = {KEY_SCA, KEY_SCA, IN_SCA};

#pragma unroll
  for (int p = 0; p < 3; ++p) {
    const GVParams pr = P[p];
    const int OS = OSd[p];
    __builtin_prefetch(pr.wsca, 0, 1);

    // ---- 1. vh[c] = vec @ w_vec1^T  (16 x 64, K = 64, per coord) ----
#pragma unroll
    for (int c = 0; c < 3; ++c) {
      for (int ot = 0; ot < 4; ++ot) {
        v8f acc = {};
#pragma unroll
        for (int kt = 0; kt < 16; ++kt) {
          const int k0 = kt * 4 + 2 * h;
          v2f a; a.x = vec_row[(k0 + 0) * 3 + c]; a.y = vec_row[(k0 + 1) * 3 + c];
          const float* wp = pr.wv1 + (ot * 16 + m) * 64 + k0;
          v2f b; b.x = wp[0]; b.y = wp[1];
          acc = wmma4(a, b, acc);
        }
#pragma unroll
        for (int r = 0; r < 8; ++r) vhL[c][r + 8 * h][ot * 16 + m] = acc[r];
      }
    }
    __syncthreads();

    // ---- 2. vnorm ----
#pragma unroll
    for (int t = 0; t < 32; ++t) {
      int id2 = lane + 32 * t;
      int mm = id2 >> 6, oo = id2 & 63;
      float x = vhL[0][mm][oo], y = vhL[1][mm][oo], z = vhL[2][mm][oo];
      vnL[mm][oo] = sqrtf(x * x + y * y + z * z);
    }
    __syncthreads();

    // ---- 3. out_s = [vnorm | sca] @ w_sca^T  (K = 320) ----
    for (int ot = 0; ot < OS / 16; ++ot) {
      v8f acc = {};
      for (int kt = 0; kt < 80; ++kt) {
        const int k0 = kt * 4 + 2 * h;
        v2f a;
        if (k0 < 64) { a.x = vnL[m][k0];       a.y = vnL[m][k0 + 1]; }
        else         { a.x = sca_row[k0 - 64]; a.y = sca_row[k0 - 63]; }
        const float* wp = pr.wsca + (ot * 16 + m) * 320 + k0;
        v2f b; b.x = wp[0]; b.y = wp[1];
        acc = wmma4(a, b, acc);
      }
#pragma unroll
      for (int r = 0; r < 8; ++r) {
        int mm = r + 8 * h, oo = ot * 16 + m;
        osL[mm][oo] = acc[r];
        outSg[p][(size_t)(tile * 16 + mm) * OS + oo] = acc[r];
      }
    }
    __syncthreads();

    // ---- 4. gate = sigmoid(out_s @ w_gate^T + b_gate) ----
    for (int ot = 0; ot < 4; ++ot) {
      v8f acc = {};
      for (int kt = 0; kt < OS / 4; ++kt) {
        const int k0 = kt * 4 + 2 * h;
        v2f a; a.x = osL[m][k0]; a.y = osL[m][k0 + 1];
        const float* wp = pr.wg + (ot * 16 + m) * OS + k0;
        v2f b; b.x = wp[0]; b.y = wp[1];
        acc = wmma4(a, b, acc);
      }
      float bgv = pr.bg[ot * 16 + m];
#pragma unroll
      for (int r = 0; r < 8; ++r) {
        float g = 1.0f / (1.0f + __expf(-(acc[r] + bgv)));
        gtL[r + 8 * h][ot * 16 + m] = g;
      }
    }
    __syncthreads();

    // ---- 5. out_v = gate * (vh @ w_vec2^T) ----
#pragma unroll
    for (int c = 0; c < 3; ++c) {
      for (int ot = 0; ot < 4; ++ot) {
        v8f acc = {};
#pragma unroll
        for (int kt = 0; kt < 16; ++kt) {
          const int k0 = kt * 4 + 2 * h;
          v2f a; a.x = vhL[c][m][k0]; a.y = vhL[c][m][k0 + 1];
          const float* wp = pr.wv2 + (ot * 16 + m) * 64 + k0;
          v2f b; b.x = wp[0]; b.y = wp[1];
          acc = wmma4(a, b, acc);
        }
#pragma unroll
        for (int r = 0; r < 8; ++r) {
          int mm = r + 8 * h, oo = ot * 16 + m;
          outVg[p][(size_t)(tile * 16 + mm) * 192 + oo * 3 + c] =
              acc[r] * gtL[mm][oo];
        }
      }
    }
    __syncthreads();
  }
}

// =====================================================================
// Kernel 2: dense per-molecule attention (scalar + vector channels),
// residual add.  Thread = one (node, head) pair (x2 per thread).
// =====================================================================
__global__ __launch_bounds__(256) void attn_kernel(
    const float* __restrict__ hsca, const float* __restrict__ hvec,
    const int* __restrict__ idxl,
    const float* __restrict__ qs, const float* __restrict__ ksp,
    const float* __restrict__ vsp, const float* __restrict__ qv,
    const float* __restrict__ kvp, const float* __restrict__ vvp,
    float* __restrict__ os, float* __restrict__ ov) {
  const int mol = blockIdx.x;
  const int tid = threadIdx.x;
  const float rs8 = 0.35355339059327373f;   // 1/sqrt(8)
  const float rs12 = 0.28867513459481287f;  // 1/sqrt(12)
#pragma unroll
  for (int rep = 0; rep < 2; ++rep) {
    const int pair = tid + 256 * rep;  // 0..511
    const int m = pair >> 4;
    const int h = pair & 15;
    const int node = mol * NPM + m;

    float q8[8], q12[12];
    const float* qsr = qs + (size_t)node * KEY_SCA + h * 8;
#pragma unroll
    for (int d = 0; d < 8; ++d) q8[d] = qsr[d];
    const float* qvr = qv + (size_t)node * 192 + h * 12;
#pragma unroll
    for (int d = 0; d < 12; ++d) q12[d] = qvr[d];

    float ss[NPM], sv[NPM];
    float mxs = -1e30f, mxv = -1e30f;
#pragma unroll
    for (int j = 0; j < NPM; ++j) {
      const float* kr = ksp + (size_t)(mol * NPM + j) * KEY_SCA + h * 8;
      float a = 0.f;
#pragma unroll
      for (int d = 0; d < 8; ++d) a += q8[d] * kr[d];
      a *= rs8; ss[j] = a; mxs = fmaxf(mxs, a);
      const float* kvr = kvp + (size_t)(mol * NPM + j) * 192 + h * 12;
      float b = 0.f;
#pragma unroll
      for (int d = 0; d < 12; ++d) b += q12[d] * kvr[d];
      b *= rs12; sv[j] = b; mxv = fmaxf(mxv, b);
    }
    float es = 0.f, ev = 0.f;
#pragma unroll
    for (int j = 0; j < NPM; ++j) {
      ss[j] = __expf(ss[j] - mxs); es += ss[j];
      sv[j] = __expf(sv[j] - mxv); ev += sv[j];
    }
    const float ies = 1.0f / es, iev = 1.0f / ev;

    float accs[16], accv[12];
#pragma unroll
    for (int d = 0; d < 16; ++d) accs[d] = 0.f;
#pragma unroll
    for (int d = 0; d < 12; ++d) accv[d] = 0.f;
#pragma unroll
    for (int j = 0; j < NPM; ++j) {
      const float a = ss[j] * ies;
      const float b = sv[j] * iev;
      const float* vr = vsp + (size_t)(mol * NPM + j) * IN_SCA + h * 16;
#pragma unroll
      for (int d = 0; d < 16; ++d) accs[d] += a * vr[d];
      const float* vvr = vvp + (size_t)(mol * NPM + j) * 192 + h * 12;
#pragma unroll
      for (int d = 0; d < 12; ++d) accv[d] += b * vvr[d];
    }
    // residual add (LN is a separate kernel)
    const int nid = idxl[node];
    const float* sca_r = hsca + (size_t)nid * IN_SCA + h * 16;
    float* osr = os + (size_t)node * IN_SCA + h * 16;
#pragma unroll
    for (int d = 0; d < 16; ++d) osr[d] = accs[d] + sca_r[d];
    const float* vec_r = hvec + (size_t)nid * 192 + h * 12;
    float* ovr = ov + (size_t)node * 192 + h * 12;
#pragma unroll
    for (int d = 0; d < 12; ++d) ovr[d] = accv[d] + vec_r[d];
  }
}

// =====================================================================
// Kernel 3: LayerNorm (in place), one wave per node.
// =====================================================================
__global__ __launch_bounds__(256) void ln_kernel(
    float* __restrict__ os, float* __restrict__ ov,
    const float* __restrict__ lsw, const float* __restrict__ lsb,
    const float* __restrict__ lvw, const float* __restrict__ lvb) {
  const int wv = threadIdx.x >> 5;
  const int lane = threadIdx.x & 31;
  const int node = blockIdx.x * 8 + wv;

  // scalar LN over 256
  float x[8];
  float* row = os + (size_t)node * IN_SCA;
  float s = 0.f;
#pragma unroll
  for (int r = 0; r < 8; ++r) { x[r] = row[lane + 32 * r]; s += x[r]; }
  const float mu = wave_sum(s) * (1.0f / 256.0f);
  float v = 0.f;
#pragma unroll
  for (int r = 0; r < 8; ++r) { float d = x[r] - mu; v += d * d; }
  const float inv = rsqrtf(wave_sum(v) * (1.0f / 256.0f) + 1e-5f);
#pragma unroll
  for (int r = 0; r < 8; ++r) {
    int d = lane + 32 * r;
    row[d] = (x[r] - mu) * inv * lsw[d] + lsb[d];
  }

  // vector LN over 64*3 = 192
  float y[6];
  float* vrow = ov + (size_t)node * 192;
  float s2 = 0.f;
#pragma unroll
  for (int r = 0; r < 6; ++r) { y[r] = vrow[lane + 32 * r]; s2 += y[r]; }
  const float mu2 = wave_sum(s2) * (1.0f / 192.0f);
  float v2 = 0.f;
#pragma unroll
  for (int r = 0; r < 6; ++r) { float d = y[r] - mu2; v2 += d * d; }
  const float inv2 = rsqrtf(wave_sum(v2) * (1.0f / 192.0f) + 1e-5f);
#pragma unroll
  for (int r = 0; r < 6; ++r) {
    int d = lane + 32 * r;
    vrow[d] = (y[r] - mu2) * inv2 * lvw[d] + lvb[d];
  }
}

// =====================================================================
// Kernel 4: per-node edge-GVLinear precompute.
//   A_n = Wv1e[:, :64] @ out_v[n],  B_n = Wv1e[:, 64:] @ out_v[n]  (128x3)
//   Si_n = Wse[:,128:384] @ out_s[n],  Sj_n = Wse[:,384:640] @ out_s[n]
// One wave per 16-node tile; GEMMs via f32 WMMA.
// =====================================================================
__global__ __launch_bounds__(32) void edge_pre_kernel(
    const float* __restrict__ os, const float* __restrict__ ov,
    const float* __restrict__ Wv1e, const float* __restrict__ Wse,
    float* __restrict__ eA, float* __restrict__ eB,
    float* __restrict__ Si, float* __restrict__ Sj) {
  const int tile = blockIdx.x;
  const int lane = threadIdx.x & 31;
  const int m = lane & 15;
  const int h = lane >> 4;

  const float* ov_row = ov + (size_t)(tile * 16 + m) * 192;

#pragma unroll
  for (int half = 0; half < 2; ++half) {
    float* dst = half ? eB : eA;
#pragma unroll
    for (int c = 0; c < 3; ++c) {
      for (int ot = 0; ot < 8; ++ot) {  // 128 output channels
        v8f acc = {};
#pragma unroll
        for (int kt = 0; kt < 16; ++kt) {
          const int k0 = kt * 4 + 2 * h;
          v2f a; a.x = ov_row[(k0 + 0) * 3 + c]; a.y = ov_row[(k0 + 1) * 3 + c];
          const float* wp = Wv1e + (ot * 16 + m) * 128 + half * 64 + k0;
          v2f b; b.x = wp[0]; b.y = wp[1];
          acc = wmma4(a, b, acc);
        }
#pragma unroll
        for (int r = 0; r < 8; ++r)
          dst[(size_t)(tile * 16 + r + 8 * h) * 384 + (ot * 16 + m) * 3 + c] =
              acc[r];
      }
    }
  }

  // Si / Sj : 16 nodes x 4 outputs x 2 matrices = 128 dots of length 256
#pragma unroll
  for (int rep = 0; rep < 4; ++rep) {
    const int id = lane + 32 * rep;  // 0..127
    const int mat = id >> 6;
    const int t = (id >> 4) & 3;
    const int m2 = id & 15;
    const float* orow = os + (size_t)(tile * 16 + m2) * IN_SCA;
    const float* w = Wse + t * 640 + 128 + mat * 256;
    float acc = 0.f;
    for (int d = 0; d < 256; ++d) acc += orow[d] * w[d];
    (mat ? Sj : Si)[(size_t)(tile * 16 + m2) * 4 + t] = acc;
  }
}

// =====================================================================
// Kernel 5: edge outputs.  One block per molecule; A/B tiles staged in
// LDS by the Tensor Data Mover (TDM) when available.
//   out[e] = vnorm(A_i + B_j) . Wse[:, :128] + Si_i + Sj_j
// =====================================================================
__global__ __launch_bounds__(256) void edge_out_kernel(
    const float* __restrict__ eA, const float* __restrict__ eB,
    const float* __restrict__ Si, const float* __restrict__ Sj,
    const float* __restrict__ Wse, const int* __restrict__ eidx,
    float* __restrict__ out) {
  __shared__ float Als[NPM][384];
  __shared__ float Bls[NPM][384];
  __shared__ float Wls[512];  // Wse[t][0..127] for t=0..3
  __shared__ float SiL[NPM][4];
  __shared__ float SjL[NPM][4];

  const int mol = blockIdx.x;
  const int tid = threadIdx.x;
  const size_t base = (size_t)mol * NPM * 384;

#ifdef HAVE_TDM
  // One DMA descriptor per 48KB tile, issued by wave 0 (TDM ignores EXEC;
  // issue is per-wave).  Other waves proceed to the small staging below and
  // meet at the barrier after wave 0 has drained TENSORcnt.
  if (tid < 32) {
    tdm_load_1d((unsigned int)(uintptr_t)&Als[0][0], eA + base, NPM * 384);
    tdm_load_1d((unsigned int)(uintptr_t)&Bls[0][0], eB + base, NPM * 384);
  }
#else
#pragma unroll
  for (int r = 0; r < 48; ++r) {
    int id = tid + 256 * r;  // 0..12287
    ((float*)Als)[id] = eA[base + id];
    ((float*)Bls)[id] = eB[base + id];
  }
#endif
#pragma unroll
  for (int r = 0; r < 2; ++r) {
    int id = tid + 256 * r;  // 0..511
    Wls[id] = Wse[(id >> 7) * 640 + (id & 127)];
  }
  if (tid < 128) {
    SiL[tid >> 2][tid & 3] = Si[(size_t)(mol * NPM) * 4 + tid];
    SjL[tid >> 2][tid & 3] = Sj[(size_t)(mol * NPM) * 4 + tid];
  }
#ifdef HAVE_TDM
  if (tid < 32) __builtin_amdgcn_s_wait_tensorcnt(0);
#endif
  __syncthreads();

  const int wv = tid >> 5;
  const int lane = tid & 31;
  for (int e = wv; e < EPM; e += 8) {
    const int ge = mol * EPM + e;
    const int li = eidx[ge] & (NPM - 1);
    const int lj = eidx[EP + ge] & (NPM - 1);
    float p0 = 0.f, p1 = 0.f, p2 = 0.f, p3 = 0.f;
#pragma unroll
    for (int r = 0; r < 4; ++r) {
      const int o = lane + 32 * r;
      const float x = Als[li][o * 3 + 0] + Bls[lj][o * 3 + 0];
      const float y = Als[li][o * 3 + 1] + Bls[lj][o * 3 + 1];
      const float z = Als[li][o * 3 + 2] + Bls[lj][o * 3 + 2];
      const float nrm = sqrtf(x * x + y * y + z * z);
      p0 += nrm * Wls[0 * 128 + o];
      p1 += nrm * Wls[1 * 128 + o];
      p2 += nrm * Wls[2 * 128 + o];
      p3 += nrm * Wls[3 * 128 + o];
    }
    p0 = wave_sum(p0); p1 = wave_sum(p1);
    p2 = wave_sum(p2); p3 = wave_sum(p3);
    if (lane == 0) {
      out[(size_t)ge * 4 + 0] = p0 + SiL[li][0] + SjL[lj][0];
      out[(size_t)ge * 4 + 1] = p1 + SiL[li][1] + SjL[lj][1];
      out[(size_t)ge * 4 + 2] = p2 + SiL[li][2] + SjL[lj][2];
      out[(size_t)ge * 4 + 3] = p3 + SiL[li][3] + SjL[lj][3];
    }
  }
}

// =====================================================================
// Host launcher.  Input order follows setup_inputs() insertion order
// with params flattened in insertion order:
//  0 h_compose_sca  1 h_compose_vec
//  2..6   q:  w_vec1 w_sca w_vec2 w_gate b_gate
//  7..11  k:  (same)   12..16 v: (same)   17..21 edge: (same)
//  22 ln_sca_w 23 ln_sca_b 24 ln_vec_w 25 ln_vec_b
//  26 idx_ligand 27 index_edge_i 28 index_edge_j 29 edge_index
// =====================================================================
extern "C" void kernel_launch(void* const* d_in, const int* in_sizes, int n_in,
                              void* d_out, int out_size, void* d_ws,
                              size_t ws_size, hipStream_t stream) {
  (void)in_sizes; (void)n_in; (void)out_size; (void)ws_size;
  const float* hsca = (const float*)d_in[0];
  const float* hvec = (const float*)d_in[1];
  GVParams pq{(const float*)d_in[2], (const float*)d_in[3],
              (const float*)d_in[4], (const float*)d_in[5],
              (const float*)d_in[6]};
  GVParams pk{(const float*)d_in[7], (const float*)d_in[8],
              (const float*)d_in[9], (const float*)d_in[10],
              (const float*)d_in[11]};
  GVParams pv{(const float*)d_in[12], (const float*)d_in[13],
              (const float*)d_in[14], (const float*)d_in[15],
              (const float*)d_in[16]};
  const float* e_wv1 = (const float*)d_in[17];
  const float* e_wsca = (const float*)d_in[18];
  const float* lsw = (const float*)d_in[22];
  const float* lsb = (const float*)d_in[23];
  const float* lvw = (const float*)d_in[24];
  const float* lvb = (const float*)d_in[25];
  const int* idxl = (const int*)d_in[26];
  const int* eidx = (const int*)d_in[29];  // (2, EP)

  float* ws = (float*)d_ws;
  float* qs = ws;                       // 8192*128
  float* ksp = qs + (size_t)N_LIG * 128;
  float* vsp = ksp + (size_t)N_LIG * 128;    // 8192*256
  float* qv = vsp + (size_t)N_LIG * 256;     // 8192*192
  float* kvp = qv + (size_t)N_LIG * 192;
  float* vvp = kvp + (size_t)N_LIG * 192;
  float* os = vvp + (size_t)N_LIG * 192;     // 8192*256
  float* ov = os + (size_t)N_LIG * 256;      // 8192*192
  float* eA = ov + (size_t)N_LIG * 192;      // 8192*384
  float* eB = eA + (size_t)N_LIG * 384;
  float* Si = eB + (size_t)N_LIG * 384;      // 8192*4
  float* Sj = Si + (size_t)N_LIG * 4;

  gv_qkv_kernel<<<N_LIG / 16, 32, 0, stream>>>(hsca, hvec, idxl, pq, pk, pv,
                                               qs, ksp, vsp, qv, kvp, vvp);
  attn_kernel<<<MOLS, 256, 0, stream>>>(hsca, hvec, idxl, qs, ksp, vsp, qv,
                                        kvp, vvp, os, ov);
  ln_kernel<<<N_LIG / 8, 256, 0, stream>>>(os, ov, lsw, lsb, lvw, lvb);
  edge_pre_kernel<<<N_LIG / 16, 32, 0, stream>>>(os, ov, e_wv1, e_wsca, eA, eB,
                                                 Si, Sj);
  edge_out_kernel<<<MOLS, 256, 0, stream>>>(eA, eB, Si, Sj, e_wsca, eidx,
                                            (float*)d_out);
}